// Encoder_18708877541797
// MI455X (gfx1250) — compile-verified
//
#include <hip/hip_runtime.h>

typedef float v2f __attribute__((ext_vector_type(2)));
typedef float v8f __attribute__((ext_vector_type(8)));

#define D 64  // feature dim (D_IN == D_HID == 64)

// Hardware fp32 atomic add (CDNA5 VGLOBAL op 86: global_atomic_add_f32).
// Inline asm guarantees the single-instruction L2 atomic path (no CAS loop).
__device__ __forceinline__ void gatomic_add_f32(float* addr, float v) {
    asm volatile("global_atomic_add_f32 %0, %1, off" : : "v"(addr), "v"(v) : "memory");
}

// ---------------------------------------------------------------------------
// Edge scatter (the hot kernel): acc[dst] += h[src] * norm[e].
// 16 lanes per edge; each lane owns one float4 (16B) -> the gather and the
// atomic adds each touch one contiguous 256B line per edge.
__global__ __launch_bounds__(256) void k_scatter(const int* __restrict__ src,
                                                 const int* __restrict__ dst,
                                                 const float* __restrict__ norm,
                                                 const float* __restrict__ H,
                                                 float* __restrict__ acc, int e_cnt) {
    const unsigned g = blockIdx.x * 256u + threadIdx.x;   // < E*16 = 25.6M
    const unsigned e = g >> 4;
    const unsigned l = g & 15u;
    if (e >= (unsigned)e_cnt) return;
    const unsigned s   = (unsigned)src[e];
    const unsigned d   = (unsigned)dst[e];
    const float    nrm = norm[e];
    const float4 hv = *(const float4*)(H + s * (unsigned)D + l * 4u);
    float* base = acc + d * (unsigned)D + l * 4u;
    gatomic_add_f32(base + 0, hv.x * nrm);
    gatomic_add_f32(base + 1, hv.y * nrm);
    gatomic_add_f32(base + 2, hv.z * nrm);
    gatomic_add_f32(base + 3, hv.w * nrm);
}

// ---------------------------------------------------------------------------
// deg[i] = 1.0 (self loop)
__global__ __launch_bounds__(256) void k_init_deg(float* deg, int n) {
    int i = blockIdx.x * blockDim.x + threadIdx.x;
    if (i < n) deg[i] = 1.0f;
}

// deg[dst[e]] += 1
__global__ __launch_bounds__(256) void k_deg_count(const int* __restrict__ dst,
                                                   float* __restrict__ deg, int e_cnt) {
    int e = blockIdx.x * blockDim.x + threadIdx.x;
    if (e < e_cnt) gatomic_add_f32(&deg[dst[e]], 1.0f);
}

// dinv[i] = rsqrt(deg[i])   (deg >= 1 always due to self loop)
__global__ __launch_bounds__(256) void k_rsqrt(float* d, int n) {
    int i = blockIdx.x * blockDim.x + threadIdx.x;
    if (i < n) d[i] = rsqrtf(d[i]);
}

// norm[e] = dinv[src[e]] * dinv[dst[e]]   (shared by all 3 layers)
__global__ __launch_bounds__(256) void k_edge_norm(const int* __restrict__ src,
                                                   const int* __restrict__ dst,
                                                   const float* __restrict__ dinv,
                                                   float* __restrict__ norm, int e_cnt) {
    int e = blockIdx.x * blockDim.x + threadIdx.x;
    if (e < e_cnt) norm[e] = dinv[src[e]] * dinv[dst[e]];
}

// acc[i] = h[i] * dinv[node]^2 + bias[c]   (self-loop term + bias = accumulator init)
__global__ __launch_bounds__(256) void k_selfinit(const float* __restrict__ H,
                                                  const float* __restrict__ dinv,
                                                  const float* __restrict__ bias,
                                                  float* __restrict__ acc, int n_nodes) {
    const long i = (long)blockIdx.x * blockDim.x + threadIdx.x;
    if (i >= (long)n_nodes * D) return;
    const int node = (int)(i >> 6);
    const int c    = (int)(i & (D - 1));
    const float di = dinv[node];
    acc[i] = H[i] * di * di + bias[c];
}

// ---------------------------------------------------------------------------
// H[M x 64] = act(X)[M x 64] @ W[64 x 64] using v_wmma_f32_16x16x4_f32.
// RELU template folds the previous layer's activation into the A-matrix load,
// eliminating a separate 25.6MB read+write pass.
// Block = 128 threads = 4 waves; wave w computes the 16x16 tile at
// (m = blockIdx.x*16, n = w*16). K-loop: 16 steps of 4.
// A frag (16x4 f32, 2 VGPR): lanes 0-15 -> K=k0..k0+1; lanes 16-31 -> K=k0+2..k0+3.
// B frag (4x16 f32, 2 VGPR): mirrored, col n=l16.
// C/D (16x16 f32, 8 VGPR): vgpr r -> M = r + 8*half, N = l16.
template <bool RELU>
__global__ __launch_bounds__(128) void k_gemm_wmma(const float* __restrict__ X,
                                                   const float* __restrict__ W,
                                                   float* __restrict__ H, int n_rows) {
    const int lane = threadIdx.x & 31;
    const int wave = threadIdx.x >> 5;   // n-tile 0..3
    const int half = lane >> 4;          // 0 or 1
    const int l16  = lane & 15;
    const int mt   = blockIdx.x;

    int row = mt * 16 + l16;
    if (row >= n_rows) row = n_rows - 1;       // clamp (keeps EXEC all-1s for WMMA)
    const int ncol = wave * 16 + l16;

    v8f c = {};
#pragma unroll
    for (int k0 = 0; k0 < D; k0 += 4) {
        const int ka = k0 + 2 * half;
        float2 av = *(const float2*)(X + (long)row * D + ka);
        if (RELU) { av.x = fmaxf(av.x, 0.0f); av.y = fmaxf(av.y, 0.0f); }
        v2f a, b;
        a.x = av.x;  a.y = av.y;
        b.x = W[ka * D + ncol];
        b.y = W[(ka + 1) * D + ncol];
        c = __builtin_amdgcn_wmma_f32_16x16x4_f32(false, a, false, b,
                                                  (short)0, c, false, false);
    }

    const int mbase = mt * 16 + half * 8;
#pragma unroll
    for (int r = 0; r < 8; ++r) {
        const int m = mbase + r;
        if (m < n_rows) H[(long)m * D + ncol] = c[r];
    }
}

// ---------------------------------------------------------------------------
extern "C" void kernel_launch(void* const* d_in, const int* in_sizes, int n_in,
                              void* d_out, int out_size, void* d_ws, size_t ws_size,
                              hipStream_t stream) {
    const float* x  = (const float*)d_in[0];
    const int*   ei = (const int*)d_in[1];
    const float* Wm[3] = { (const float*)d_in[2], (const float*)d_in[4], (const float*)d_in[6] };
    const float* bm[3] = { (const float*)d_in[3], (const float*)d_in[5], (const float*)d_in[7] };

    const int N = in_sizes[0] / D;        // 100000
    const int E = in_sizes[1] / 2;        // 1600000
    const int* src = ei;
    const int* dst = ei + E;

    // workspace layout (floats): dinv | norm | h | acc
    const long npad = ((long)N + 63) & ~63L;
    const long epad = ((long)E + 63) & ~63L;
    const long feat = (long)N * D;
    float* dinv = (float*)d_ws;
    float* norm = dinv + npad;
    float* h    = norm + epad;
    float* acc  = h + feat;

    const int B = 256;
    const int gN    = (N + B - 1) / B;
    const int gE    = (E + B - 1) / B;
    const int gFeat = (int)((feat + B - 1) / B);
    const int gScat = (int)(((long)E * 16 + B - 1) / B);
    const int gGemm = (N + 15) / 16;      // 6250, exact

    // normalization: deg -> dinv -> per-edge norm (reused by all 3 layers)
    k_init_deg<<<gN, B, 0, stream>>>(dinv, N);
    k_deg_count<<<gE, B, 0, stream>>>(dst, dinv, E);
    k_rsqrt<<<gN, B, 0, stream>>>(dinv, N);
    k_edge_norm<<<gE, B, 0, stream>>>(src, dst, dinv, norm, E);

    // Layer 1: h = x @ W1 ; acc = h*dinv^2 + b1 ; acc += scatter(edges)
    k_gemm_wmma<false><<<gGemm, 128, 0, stream>>>(x, Wm[0], h, N);
    k_selfinit<<<gFeat, B, 0, stream>>>(h, dinv, bm[0], acc, N);
    k_scatter<<<gScat, B, 0, stream>>>(src, dst, norm, h, acc, E);

    // Layer 2: h = relu(acc) @ W2 (relu fused into GEMM A-load) ; re-init acc
    k_gemm_wmma<true><<<gGemm, 128, 0, stream>>>(acc, Wm[1], h, N);
    k_selfinit<<<gFeat, B, 0, stream>>>(h, dinv, bm[1], acc, N);
    k_scatter<<<gScat, B, 0, stream>>>(src, dst, norm, h, acc, E);

    // Layer 3: h = relu(acc) @ W3 ; aggregate straight into d_out (no relu)
    float* out = (float*)d_out;
    k_gemm_wmma<true><<<gGemm, 128, 0, stream>>>(acc, Wm[2], h, N);
    k_selfinit<<<gFeat, B, 0, stream>>>(h, dinv, bm[2], out, N);
    k_scatter<<<gScat, B, 0, stream>>>(src, dst, norm, h, out, E);
}